// InterpolationDecoder_76965813944562
// MI455X (gfx1250) — compile-verified
//
#include <hip/hip_runtime.h>
#include <hip/hip_bf16.h>

// Problem constants (match reference)
#define QN   4096
#define NN   1024
#define FDIM 64
#define HDIM 128
#define TQ   16      // queries per workgroup
#define NCHUNK 128   // bn rows staged in LDS per chunk
#define NCHUNKS (NN / NCHUNK)

typedef float v2f __attribute__((ext_vector_type(2)));
typedef float v8f __attribute__((ext_vector_type(8)));

// V_WMMA_F32_16X16X4_F32: D(16x16 f32) = A(16x4 f32) * B(4x16 f32) + C
// A/B fragments: 2 VGPRs/lane. lanes 0-15 hold K={0,1}, lanes 16-31 hold K={2,3}.
// C/D: VGPR r -> (M=r, N=lane) for lanes 0-15, (M=r+8, N=lane-16) for lanes 16-31.
__device__ __forceinline__ v8f wmma_f32_k4(v2f a, v2f b, v8f c) {
    return __builtin_amdgcn_wmma_f32_16x16x4_f32(
        /*neg_a=*/false, a, /*neg_b=*/false, b,
        /*c_mod=*/(short)0, c, /*reuse_a=*/false, /*reuse_b=*/false);
}

// ---- CDNA5 async global->LDS (GLOBAL_LOAD_ASYNC_TO_LDS_B128, ASYNCcnt) ----
// Probe-confirmed (round-2 diagnostic): param 0 is `int __vector(4) AS(1)*`,
// i.e. signature  (v4i global*, v4i shared*, imm offset, imm cpol).
typedef int i32x4 __attribute__((vector_size(4 * sizeof(int))));

#if __has_builtin(__builtin_amdgcn_global_load_async_to_lds_b128)
#define HAVE_ASYNC_LDS 1
__device__ __forceinline__ void async_copy16(const float* g, float* l) {
    __builtin_amdgcn_global_load_async_to_lds_b128(
        (__attribute__((address_space(1))) i32x4*)(const_cast<float*>(g)),
        (__attribute__((address_space(3))) i32x4*)(l),
        /*imm offset=*/0, /*cpol=*/0);
}
__device__ __forceinline__ void async_wait0() {
#if __has_builtin(__builtin_amdgcn_s_wait_asynccnt)
    __builtin_amdgcn_s_wait_asynccnt(0);
#else
    asm volatile("s_wait_asynccnt 0x0" ::: "memory");
#endif
}
#else
#define HAVE_ASYNC_LDS 0
__device__ __forceinline__ void async_copy16(const float* g, float* l) {
    *reinterpret_cast<float4*>(l) = *reinterpret_cast<const float4*>(g);
}
__device__ __forceinline__ void async_wait0() {}
#endif

// ---------------------------------------------------------------------------
// Kernel 1: bn[n,h] = (nf @ aW1[:F])[n,h] + ab1[h] - npos[n,:] . aW1[F:,h]
// WMMA GEMM with the query-independent position term folded into the epilogue.
// One wave per 16x16 tile. 64 n-tiles x 8 h-tiles = 512 waves.
// ---------------------------------------------------------------------------
__global__ __launch_bounds__(128) void bn_gemm_kernel(
    const float* __restrict__ nf,    // [N, F]
    const float* __restrict__ aW1,   // [F+3, H]
    const float* __restrict__ ab1,   // [H]
    const float* __restrict__ npos,  // [N, 3]
    float* __restrict__ bn)          // [N, H]
{
    const int wave = (blockIdx.x * blockDim.x + threadIdx.x) >> 5;
    const int lane = threadIdx.x & 31;
    const int nt = wave >> 3;          // 0..63
    const int ht = wave & 7;           // 0..7
    const int n0 = nt * 16, h0 = ht * 16;
    const int lm = lane & 15;          // M (A) / N (B,D) index
    const int hi = lane >> 4;          // selects K pair {0,1} vs {2,3}

    v8f c = {};
    #pragma unroll
    for (int k0 = 0; k0 < FDIM; k0 += 4) {
        const int k = k0 + hi * 2;
        v2f a, b;
        a.x = nf[(n0 + lm) * FDIM + k + 0];
        a.y = nf[(n0 + lm) * FDIM + k + 1];
        b.x = aW1[(k + 0) * HDIM + h0 + lm];
        b.y = aW1[(k + 1) * HDIM + h0 + lm];
        c = wmma_f32_k4(a, b, c);
    }
    const float wx = aW1[(FDIM + 0) * HDIM + h0 + lm];
    const float wy = aW1[(FDIM + 1) * HDIM + h0 + lm];
    const float wz = aW1[(FDIM + 2) * HDIM + h0 + lm];
    const float bias = ab1[h0 + lm];
    #pragma unroll
    for (int r = 0; r < 8; ++r) {
        const int n = n0 + r + hi * 8;
        const float px = npos[n * 3 + 0];
        const float py = npos[n * 3 + 1];
        const float pz = npos[n * 3 + 2];
        bn[n * HDIM + h0 + lm] =
            c[r] + bias - px * wx - py * wy - pz * wz;
    }
}

// ---------------------------------------------------------------------------
// Kernel 2: fused per-query-tile pipeline.
//   scores (VALU, async double-buffered bn) -> softmax -> agg (WMMA, 8-wave
//   K-split) -> decoder L1 (WMMA) -> decoder L2 -> out
// 256 threads (8 wave32) per block, one block per 16 queries.
// ---------------------------------------------------------------------------
#define SROW  (NN + 1)      // sbuf row stride (bank spread for WMMA A reads)
#define BROW  (HDIM + 4)    // basec row stride (16B aligned + bank spread)
#define AROW  (FDIM + 4)    // aggl row stride
#define HROW  (HDIM + 4)    // bqs/hid row stride (16B aligned)

__global__ __launch_bounds__(256) void fused_attn_kernel(
    const float* __restrict__ qp,    // [Q,3]
    const float* __restrict__ nf,    // [N,F]
    const float* __restrict__ npos,  // [N,3]
    const float* __restrict__ aW1,   // [F+3,H]
    const float* __restrict__ aW2,   // [H,1]
    const float* __restrict__ ab2,   // [1]
    const float* __restrict__ dW1,   // [F,H]
    const float* __restrict__ db1,   // [H]
    const float* __restrict__ dW2,   // [H,3]
    const float* __restrict__ db2,   // [3]
    const float* __restrict__ bn,    // [N,H] (from kernel 1)
    float* __restrict__ out)         // [Q,3]
{
    __shared__ float sbuf[TQ][SROW];           // scores -> attn (64KB)
    __shared__ float basec[2][NCHUNK][BROW];   // double-buffered bn chunks
    __shared__ float bqs[TQ][HROW];            // query-side hidden offsets
    __shared__ float a2s[HDIM];                // aW2
    __shared__ float qpos[TQ][2];              // query xy
    __shared__ float nposa[NN][2];             // node xy
    __shared__ float red[TQ][16];
    __shared__ float red2[TQ];
    __shared__ float aggl[TQ][AROW];
    __shared__ float hid[TQ][HROW];

    const int tid  = threadIdx.x;
    const int q0   = blockIdx.x * TQ;
    const int wave = tid >> 5;
    const int lane = tid & 31;
    const int lm   = lane & 15;
    const int hi   = lane >> 4;

    // ---- Phase 0: stage small operands --------------------------------------
    if (tid < HDIM) a2s[tid] = aW2[tid];
    if (tid < TQ) {
        qpos[tid][0] = qp[(q0 + tid) * 3 + 0];
        qpos[tid][1] = qp[(q0 + tid) * 3 + 1];
    }
    for (int i = tid; i < NN * 2; i += 256)
        (&nposa[0][0])[i] = npos[(i >> 1) * 3 + (i & 1)];
    // bq[q,h] = qp[q,:] . aW1[F:,h]  (reads globals only -> no ordering dep)
    for (int i = tid; i < TQ * HDIM; i += 256) {
        const int q = i >> 7, h = i & (HDIM - 1);
        const float qx = qp[(q0 + q) * 3 + 0];
        const float qy = qp[(q0 + q) * 3 + 1];
        const float qz = qp[(q0 + q) * 3 + 2];
        bqs[q][h] = qx * aW1[(FDIM + 0) * HDIM + h]
                  + qy * aW1[(FDIM + 1) * HDIM + h]
                  + qz * aW1[(FDIM + 2) * HDIM + h];
    }
    const float ab2v = ab2[0];

    // kick off async stage of chunk 0 while Phase 0 finishes
    auto stage = [&](int c, int buf) {
        const float* src = bn + (c * NCHUNK) * HDIM;
        for (int i = tid; i < NCHUNK * (HDIM / 4); i += 256) {
            const int row = i >> 5;            // HDIM/4 == 32
            const int c4  = i & 31;
            async_copy16(src + row * HDIM + c4 * 4, &basec[buf][row][c4 * 4]);
        }
    };
    stage(0, 0);
    __syncthreads();   // Phase-0 LDS visible

    // ---- Phase 1: pairwise scores, s[q,n] = (sum_h relu(bn+bq)*a2 + ab2)*w --
    const int sq = tid >> 4;   // query within tile (0..15)
    const int ss = tid & 15;   // n-slot (0..15)
    const float qx = qpos[sq][0], qy = qpos[sq][1];

    for (int c = 0; c < NCHUNKS; ++c) {
        async_wait0();      // this wave's async chunk writes complete
        __syncthreads();    // => all waves' writes complete: buffer c&1 ready
        if (c + 1 < NCHUNKS) stage(c + 1, (c + 1) & 1);  // prefetch other buf

        const int bsel = c & 1;
        float acc[8];
        #pragma unroll
        for (int j = 0; j < 8; ++j) acc[j] = 0.0f;

        #pragma unroll 2
        for (int h4 = 0; h4 < HDIM; h4 += 4) {
            const float4 bq4 = *reinterpret_cast<const float4*>(&bqs[sq][h4]);
            const float4 a24 = *reinterpret_cast<const float4*>(&a2s[h4]);
            #pragma unroll
            for (int j = 0; j < 8; ++j) {
                const int nl = ss + 16 * j;
                const float4 b4 =
                    *reinterpret_cast<const float4*>(&basec[bsel][nl][h4]);
                acc[j] = fmaf(fmaxf(b4.x + bq4.x, 0.0f), a24.x, acc[j]);
                acc[j] = fmaf(fmaxf(b4.y + bq4.y, 0.0f), a24.y, acc[j]);
                acc[j] = fmaf(fmaxf(b4.z + bq4.z, 0.0f), a24.z, acc[j]);
                acc[j] = fmaf(fmaxf(b4.w + bq4.w, 0.0f), a24.w, acc[j]);
            }
        }
        #pragma unroll
        for (int j = 0; j < 8; ++j) {
            const int n = c * NCHUNK + ss + 16 * j;
            const float dx = qx - nposa[n][0];
            const float dy = qy - nposa[n][1];
            const float w = 1.0f / (sqrtf(dx * dx + dy * dy) + 1e-6f);
            sbuf[sq][n] = (acc[j] + ab2v) * w;
        }
        // no trailing barrier needed: prefetch targets the other buffer and
        // next iteration begins with asynccnt-wait + barrier.
    }
    __syncthreads();

    // ---- Phase 2: softmax over n (16 slots x 64 elements per query) ---------
    {
        const int q = tid & 15;
        const int slot = tid >> 4;
        const int nbeg = slot * (NN / 16);

        float m = -3.4e38f;
        #pragma unroll 8
        for (int j = 0; j < NN / 16; ++j) m = fmaxf(m, sbuf[q][nbeg + j]);
        red[q][slot] = m;
        __syncthreads();
        if (slot == 0) {
            float mm = red[q][0];
            #pragma unroll
            for (int s = 1; s < 16; ++s) mm = fmaxf(mm, red[q][s]);
            red2[q] = mm;
        }
        __syncthreads();
        const float mv = red2[q];
        float sum = 0.0f;
        #pragma unroll 8
        for (int j = 0; j < NN / 16; ++j) {
            const float e = __expf(sbuf[q][nbeg + j] - mv);
            sbuf[q][nbeg + j] = e;
            sum += e;
        }
        red[q][slot] = sum;
        __syncthreads();
        if (slot == 0) {
            float t = 0.0f;
            #pragma unroll
            for (int s = 0; s < 16; ++s) t += red[q][s];
            red2[q] = 1.0f / t;
        }
        __syncthreads();
        const float inv = red2[q];
        #pragma unroll 8
        for (int j = 0; j < NN / 16; ++j) sbuf[q][nbeg + j] *= inv;
    }
    __syncthreads();

    // ---- Phase 3: agg[16,64] = attn[16,1024] @ nf[1024,64] ------------------
    // WMMA, all 8 waves: 4 f-tiles x 2-way K-split, LDS reduction.
    {
        const int ft = wave & 3, kh = wave >> 2;
        const int f0 = ft * 16;
        v8f c = {};
        const int kbeg = kh * (NN / 2), kend = kbeg + NN / 2;
        for (int k0 = kbeg; k0 < kend; k0 += 4) {
            const int k = k0 + hi * 2;
            v2f a, b;
            a.x = sbuf[lm][k + 0];
            a.y = sbuf[lm][k + 1];
            b.x = nf[(k + 0) * FDIM + f0 + lm];
            b.y = nf[(k + 1) * FDIM + f0 + lm];
            c = wmma_f32_k4(a, b, c);
        }
        if (kh == 0) {
            #pragma unroll
            for (int r = 0; r < 8; ++r)
                aggl[r + hi * 8][f0 + lm] = c[r];
        }
        __syncthreads();
        if (kh == 1) {
            #pragma unroll
            for (int r = 0; r < 8; ++r)
                aggl[r + hi * 8][f0 + lm] += c[r];
        }
    }
    __syncthreads();

    // ---- Phase 4: hid = relu(agg @ dW1 + db1)  (WMMA, 8 h-tiles) ------------
    {
        const int h0 = wave * 16;
        v8f c = {};
        #pragma unroll
        for (int k0 = 0; k0 < FDIM; k0 += 4) {
            const int k = k0 + hi * 2;
            v2f a, b;
            a.x = aggl[lm][k + 0];
            a.y = aggl[lm][k + 1];
            b.x = dW1[(k + 0) * HDIM + h0 + lm];
            b.y = dW1[(k + 1) * HDIM + h0 + lm];
            c = wmma_f32_k4(a, b, c);
        }
        const float bv = db1[h0 + lm];
        #pragma unroll
        for (int r = 0; r < 8; ++r)
            hid[r + hi * 8][h0 + lm] = fmaxf(c[r] + bv, 0.0f);
    }
    __syncthreads();

    // ---- Phase 5: out = hid @ dW2 + db2  (tiny, VALU) -----------------------
    if (tid < TQ * 3) {
        const int q = tid / 3, o = tid % 3;
        float acc = db2[o];
        #pragma unroll 4
        for (int h = 0; h < HDIM; ++h)
            acc = fmaf(hid[q][h], dW2[h * 3 + o], acc);
        out[(q0 + q) * 3 + o] = acc;
    }
}

// ---------------------------------------------------------------------------
extern "C" void kernel_launch(void* const* d_in, const int* in_sizes, int n_in,
                              void* d_out, int out_size, void* d_ws, size_t ws_size,
                              hipStream_t stream) {
    const float* qp   = (const float*)d_in[0];   // [Q,3]
    const float* nf   = (const float*)d_in[1];   // [N,F]
    const float* npos = (const float*)d_in[2];   // [N,3]
    const float* aW1  = (const float*)d_in[3];   // [F+3,H]
    const float* ab1  = (const float*)d_in[4];   // [H]
    const float* aW2  = (const float*)d_in[5];   // [H,1]
    const float* ab2  = (const float*)d_in[6];   // [1]
    const float* dW1  = (const float*)d_in[7];   // [F,H]
    const float* db1  = (const float*)d_in[8];   // [H]
    const float* dW2  = (const float*)d_in[9];   // [H,3]
    const float* db2  = (const float*)d_in[10];  // [3]
    float* out = (float*)d_out;

    float* bn = (float*)d_ws;                    // [N,H] = 512 KB scratch

    // 512 tile-waves, 4 waves per block -> 128 blocks of 128 threads
    bn_gemm_kernel<<<128, 128, 0, stream>>>(nf, aW1, ab1, npos, bn);

    // one block per 16 queries
    fused_attn_kernel<<<QN / TQ, 256, 0, stream>>>(
        qp, nf, npos, aW1, aW2, ab2, dW1, db1, dW2, db2, bn, out);
}